// Net_61160334295401
// MI455X (gfx1250) — compile-verified
//
#include <hip/hip_runtime.h>
#include <math.h>

// Problem constants (hard-coded in the reference source)
#define FDIM 256
#define HDIM 5
#define CDIM 7

typedef __attribute__((ext_vector_type(2))) float v2f;
typedef __attribute__((ext_vector_type(8))) float v8f;

// ---------------------------------------------------------------------------
// Stage 1: degree counting (integer atomics into L2) and deg^-1/2
// ---------------------------------------------------------------------------
__global__ void deg_count_kernel(const int* __restrict__ dst,
                                 int* __restrict__ cnt, int e) {
    int i = blockIdx.x * blockDim.x + threadIdx.x;
    if (i < e) atomicAdd(&cnt[dst[i]], 1);
}

__global__ void deg_inv_kernel(const int* __restrict__ cnt,
                               float* __restrict__ dis, int n) {
    int i = blockIdx.x * blockDim.x + threadIdx.x;
    if (i < n) dis[i] = rsqrtf((float)cnt[i] + 1.0f);  // +1 for self-loop
}

// ---------------------------------------------------------------------------
// Stage 2: h1 = x @ W1 via V_WMMA_F32_16X16X4_F32.
// One wave owns a 16-row tile of x; K loop of 64 WMMA steps (K=4 each).
// W1 (256x5) is zero-padded to 256x16 in LDS so B loads are clean ds_loads.
//
// Operand layout (ISA 7.12.2, 32-bit A 16x4): lane L holds row L%16;
// VGPR0/1 carry K = {0,1} for lanes 0-15 and K = {2,3} for lanes 16-31.
// B (4x16) mirrors with N across lanes; D: VGPR r -> M = r + 8*(lane/16),
// N = lane%16.
// ---------------------------------------------------------------------------
__global__ void __launch_bounds__(256)
gemm1_wmma_kernel(const float* __restrict__ x, const float* __restrict__ W1,
                  float* __restrict__ h1, int n) {
    __shared__ float Wpad[FDIM * 16];  // 16 KB of the 320 KB/WGP LDS
    for (int idx = threadIdx.x; idx < FDIM * 16; idx += 256) {
        int k = idx >> 4, c = idx & 15;
        Wpad[idx] = (c < HDIM) ? W1[k * HDIM + c] : 0.0f;
    }
    __syncthreads();

    const int lane = threadIdx.x & 31;
    const int wave = threadIdx.x >> 5;
    const int tile = blockIdx.x * 8 + wave;            // 8 waves per block
    const int ntiles = (n + 15) >> 4;
    if (tile >= ntiles) return;                        // wave-uniform: EXEC stays all-1s

    const int row0 = tile * 16;
    const int r    = lane & 15;                        // row (A) / col (B,D)
    const int hi   = lane >> 4;                        // lane half
    const int kb   = hi * 2;                           // K sub-offset {0,2}

    int arow = row0 + r;
    if (arow >= n) arow = n - 1;                       // safe tail clamp (stores guarded)
    const float* xrow = x + (size_t)arow * FDIM;

    v8f acc = {};
#pragma unroll 8
    for (int k = 0; k < FDIM; k += 4) {
        // A: two consecutive K values for this lane -> one 8-byte load
        v2f a = *(const v2f*)(xrow + k + kb);
        // B: padded weights from LDS, stride-16 rows -> conflict-free banks
        v2f b;
        b.x = Wpad[(k + kb) * 16 + r];
        b.y = Wpad[(k + kb + 1) * 16 + r];
        // (neg_a, A, neg_b, B, c_mod, C, reuse_a, reuse_b)
        acc = __builtin_amdgcn_wmma_f32_16x16x4_f32(
            false, a, false, b, (short)0, acc, false, false);
    }

    // D -> compact [n, HDIM] store; only the first HDIM columns are real
    if (r < HDIM) {
#pragma unroll
        for (int r8 = 0; r8 < 8; ++r8) {
            int row = row0 + r8 + hi * 8;
            if (row < n) h1[(size_t)row * HDIM + r] = acc[r8];
        }
    }
}

// ---------------------------------------------------------------------------
// Stage 3: per-edge normalized scatter-add: agg[dst] += h[src]*dis[src]*dis[dst]
// Target arrays are 2 MB -> resident in L2; float atomics resolve on-chip.
// ---------------------------------------------------------------------------
__global__ void aggregate_kernel(const int* __restrict__ src,
                                 const int* __restrict__ dst,
                                 const float* __restrict__ dis,
                                 const float* __restrict__ h,
                                 float* __restrict__ agg, int e) {
    int i = blockIdx.x * blockDim.x + threadIdx.x;
    if (i >= e) return;
    int s = src[i], d = dst[i];
    float norm = dis[s] * dis[d];
#pragma unroll
    for (int j = 0; j < HDIM; ++j)
        atomicAdd(&agg[(size_t)d * HDIM + j], h[(size_t)s * HDIM + j] * norm);
}

// ---------------------------------------------------------------------------
// Stage 4: finalize layer 1 (self-loop + bias + relu) fused with 5x5 matvec
// ---------------------------------------------------------------------------
__global__ void finalize1_kernel(const float* __restrict__ agg,
                                 const float* __restrict__ hlin,
                                 const float* __restrict__ dis,
                                 const float* __restrict__ b1,
                                 const float* __restrict__ W2,
                                 float* __restrict__ h2, int n) {
    int i = blockIdx.x * blockDim.x + threadIdx.x;
    if (i >= n) return;
    float di2 = dis[i] * dis[i];
    float v[HDIM];
#pragma unroll
    for (int j = 0; j < HDIM; ++j) {
        float t = agg[(size_t)i * HDIM + j] + hlin[(size_t)i * HDIM + j] * di2 + b1[j];
        v[j] = t > 0.0f ? t : 0.0f;
    }
#pragma unroll
    for (int c = 0; c < HDIM; ++c) {
        float s = 0.0f;
#pragma unroll
        for (int j = 0; j < HDIM; ++j) s += v[j] * W2[j * HDIM + c];
        h2[(size_t)i * HDIM + c] = s;
    }
}

// ---------------------------------------------------------------------------
// Stage 5: finalize layer 2 + classifier head -> d_out [n, CDIM]
// ---------------------------------------------------------------------------
__global__ void finalize2_kernel(const float* __restrict__ agg,
                                 const float* __restrict__ hlin,
                                 const float* __restrict__ dis,
                                 const float* __restrict__ b2,
                                 const float* __restrict__ W3,
                                 const float* __restrict__ b3,
                                 float* __restrict__ out, int n) {
    int i = blockIdx.x * blockDim.x + threadIdx.x;
    if (i >= n) return;
    float di2 = dis[i] * dis[i];
    float v[HDIM];
#pragma unroll
    for (int j = 0; j < HDIM; ++j) {
        float t = agg[(size_t)i * HDIM + j] + hlin[(size_t)i * HDIM + j] * di2 + b2[j];
        v[j] = t > 0.0f ? t : 0.0f;
    }
#pragma unroll
    for (int c = 0; c < CDIM; ++c) {
        float s = b3[c];
#pragma unroll
        for (int j = 0; j < HDIM; ++j) s += v[j] * W3[j * CDIM + c];
        out[(size_t)i * CDIM + c] = s;
    }
}

// ---------------------------------------------------------------------------
extern "C" void kernel_launch(void* const* d_in, const int* in_sizes, int n_in,
                              void* d_out, int out_size, void* d_ws, size_t ws_size,
                              hipStream_t stream) {
    const float* x  = (const float*)d_in[0];
    const int*   ei = (const int*)d_in[1];
    const float* W1 = (const float*)d_in[2];
    const float* b1 = (const float*)d_in[3];
    const float* W2 = (const float*)d_in[4];
    const float* b2 = (const float*)d_in[5];
    const float* W3 = (const float*)d_in[6];
    const float* b3 = (const float*)d_in[7];
    float* out = (float*)d_out;

    const int n = in_sizes[0] / FDIM;     // 100,000
    const int e = in_sizes[1] / 2;        // 3,200,000
    const int* src = ei;                  // edge_index[0]
    const int* dst = ei + e;              // edge_index[1]

    // Workspace carve-out (256B-aligned regions): ~8.8 MB total
    char* ws = (char*)d_ws;
    size_t off = 0;
    auto carve = [&](size_t bytes) -> char* {
        char* p = ws + off;
        off = (off + bytes + 255) & ~(size_t)255;
        return p;
    };
    int*   cnt  = (int*)carve((size_t)4 * n);
    float* dis  = (float*)carve((size_t)4 * n);
    float* h1   = (float*)carve((size_t)4 * HDIM * n);
    float* agg1 = (float*)carve((size_t)4 * HDIM * n);
    float* h2   = (float*)carve((size_t)4 * HDIM * n);
    float* agg2 = (float*)carve((size_t)4 * HDIM * n);

    // Zero accumulation targets (graph-capture-legal async memsets)
    hipMemsetAsync(cnt,  0, (size_t)4 * n, stream);
    hipMemsetAsync(agg1, 0, (size_t)4 * HDIM * n, stream);
    hipMemsetAsync(agg2, 0, (size_t)4 * HDIM * n, stream);

    const int TB = 256;
    const int eb = (e + TB - 1) / TB;
    const int nb = (n + TB - 1) / TB;

    deg_count_kernel<<<eb, TB, 0, stream>>>(dst, cnt, e);
    deg_inv_kernel<<<nb, TB, 0, stream>>>(cnt, dis, n);

    const int ntiles = (n + 15) / 16;                 // 16 rows per wave
    gemm1_wmma_kernel<<<(ntiles + 7) / 8, 256, 0, stream>>>(x, W1, h1, n);

    aggregate_kernel<<<eb, TB, 0, stream>>>(src, dst, dis, h1, agg1, e);
    finalize1_kernel<<<nb, TB, 0, stream>>>(agg1, h1, dis, b1, W2, h2, n);
    aggregate_kernel<<<eb, TB, 0, stream>>>(src, dst, dis, h2, agg2, e);
    finalize2_kernel<<<nb, TB, 0, stream>>>(agg2, h2, dis, b2, W3, b3, out, n);
}